// HashGridEncoder_1683627180189
// MI455X (gfx1250) — compile-verified
//
#include <hip/hip_runtime.h>
#include <stdint.h>

// Hash-grid encode (Instant-NGP style) for MI455X / gfx1250.
// One thread per (point, level); wave32 covers 2 points x 16 levels so the
// 8B/thread output stores are fully coalesced. Hash table (64MB) is kept
// resident in the 192MB L2 by streaming the output with non-temporal stores.
// Block's 16 points are staged global->LDS with the CDNA5 async-to-LDS path.

typedef __attribute__((ext_vector_type(2))) float f32x2;

#define HG_NUM_LEVELS 16u
#define HG_TABLE_SIZE (1u << 19)
#define HG_TABLE_MASK (HG_TABLE_SIZE - 1u)
#define HG_PTS_PER_BLOCK 16u
#define HG_BLOCK 256u

__device__ __forceinline__ f32x2 hg_lerp2(f32x2 a, f32x2 b, float w) {
  f32x2 r;
  r.x = fmaf(w, b.x - a.x, a.x);
  r.y = fmaf(w, b.y - a.y, a.y);
  return r;
}

__global__ __launch_bounds__(HG_BLOCK) void hashgrid_encode_kernel(
    const float* __restrict__ points,   // (N, 3) f32
    const f32x2* __restrict__ table,    // (TABLE_SIZE * 16, 2) f32 as f32x2 rows
    f32x2* __restrict__ out,            // (N, 16) f32x2  == (N, 32) f32
    uint32_t n_points) {
  __shared__ float shp[HG_PTS_PER_BLOCK * 3];  // 16 points * xyz

  const uint32_t tid = threadIdx.x;
  const uint32_t pt0 = blockIdx.x * HG_PTS_PER_BLOCK;

  // --- CDNA5 async global->LDS staging of this block's 16 points (48 dwords).
  if (tid < HG_PTS_PER_BLOCK * 3u) {
    uint32_t g = pt0 * 3u + tid;
    const uint32_t gmax = n_points * 3u - 1u;
    if (g > gmax) g = gmax;  // clamp for a possible ragged last block
    const uint32_t lds =
        (uint32_t)(uintptr_t)(const __attribute__((address_space(3))) void*)&shp[tid];
    const uint64_t ga = (uint64_t)(uintptr_t)(points + g);
    asm volatile("global_load_async_to_lds_b32 %0, %1, off"
                 :: "v"(lds), "v"(ga) : "memory");
  }
  asm volatile("s_wait_asynccnt 0x0" ::: "memory");  // drain ASYNCcnt
  __syncthreads();

  const uint32_t lp = tid >> 4;      // local point 0..15
  const uint32_t level = tid & 15u;  // level 0..15 (fastest -> coalesced stores)
  const uint32_t pt = pt0 + lp;
  if (pt >= n_points) return;

  const float px = shp[lp * 3 + 0];  // LDS broadcast reads (16 lanes same addr)
  const float py = shp[lp * 3 + 1];
  const float pz = shp[lp * 3 + 2];

  // growth = exp((ln(2^19)-ln(16))/15) == 2 exactly -> res_l = 16 << l
  const float res = (float)(16u << level);

  const float sx = px * res + 0.5f;
  const float sy = py * res + 0.5f;
  const float sz = pz * res + 0.5f;
  const float fx = floorf(sx), fy = floorf(sy), fz = floorf(sz);
  const float wx = sx - fx, wy = sy - fy, wz = sz - fz;

  const uint32_t ix = (uint32_t)(int)fx;
  const uint32_t iy = (uint32_t)(int)fy;
  const uint32_t iz = (uint32_t)(int)fz;

  const uint32_t P2 = 2654435761u, P3 = 805459861u;
  const uint32_t hx0 = ix;        const uint32_t hx1 = ix + 1u;
  const uint32_t hy0 = iy * P2;   const uint32_t hy1 = hy0 + P2;
  const uint32_t hz0 = iz * P3;   const uint32_t hz1 = hz0 + P3;
  const uint32_t base = level * HG_TABLE_SIZE;

  // 8 independent 8B gathers; table stays L2-resident (regular temporal hint).
  const f32x2 f000 = table[base + ((hx0 ^ hy0 ^ hz0) & HG_TABLE_MASK)];
  const f32x2 f100 = table[base + ((hx1 ^ hy0 ^ hz0) & HG_TABLE_MASK)];
  const f32x2 f010 = table[base + ((hx0 ^ hy1 ^ hz0) & HG_TABLE_MASK)];
  const f32x2 f110 = table[base + ((hx1 ^ hy1 ^ hz0) & HG_TABLE_MASK)];
  const f32x2 f001 = table[base + ((hx0 ^ hy0 ^ hz1) & HG_TABLE_MASK)];
  const f32x2 f101 = table[base + ((hx1 ^ hy0 ^ hz1) & HG_TABLE_MASK)];
  const f32x2 f011 = table[base + ((hx0 ^ hy1 ^ hz1) & HG_TABLE_MASK)];
  const f32x2 f111 = table[base + ((hx1 ^ hy1 ^ hz1) & HG_TABLE_MASK)];

  // Trilinear blend as a lerp tree (7 fused lerps).
  const f32x2 c00 = hg_lerp2(f000, f100, wx);
  const f32x2 c10 = hg_lerp2(f010, f110, wx);
  const f32x2 c01 = hg_lerp2(f001, f101, wx);
  const f32x2 c11 = hg_lerp2(f011, f111, wx);
  const f32x2 c0  = hg_lerp2(c00, c10, wy);
  const f32x2 c1  = hg_lerp2(c01, c11, wy);
  const f32x2 r   = hg_lerp2(c0, c1, wz);

  // Streaming store: output is written once and never re-read -> keep it out
  // of L2 so the hash table stays resident (CDNA5 TH=NT store hint).
  __builtin_nontemporal_store(r, &out[(size_t)pt * HG_NUM_LEVELS + level]);
}

extern "C" void kernel_launch(void* const* d_in, const int* in_sizes, int n_in,
                              void* d_out, int out_size, void* d_ws, size_t ws_size,
                              hipStream_t stream) {
  (void)n_in; (void)out_size; (void)d_ws; (void)ws_size;
  const float* points = (const float*)d_in[0];   // (N, 3) f32
  const f32x2* table  = (const f32x2*)d_in[1];   // (2^19 * 16, 2) f32
  f32x2* out = (f32x2*)d_out;                    // (N, 32) f32

  const uint32_t n_points = (uint32_t)(in_sizes[0] / 3);
  const uint32_t blocks = (n_points + HG_PTS_PER_BLOCK - 1u) / HG_PTS_PER_BLOCK;

  hipLaunchKernelGGL(hashgrid_encode_kernel, dim3(blocks), dim3(HG_BLOCK), 0,
                     stream, points, table, out, n_points);
}